// PostHocRiemannianQuantizer_11965778886880
// MI455X (gfx1250) — compile-verified
//
#include <hip/hip_runtime.h>

// PostHocRiemannianQuantizer: fused f32 GEMM (Z @ E^T) + row-argmin.
// argmin_k g[n]*(||z_n||^2 + ||e_k||^2 - 2 z_n.e_k) == argmin_k (||e_k||^2 - 2 z_n.e_k)
// since g[n] > 0 and ||z_n||^2 are per-row constants. Exact f32 matrix path via
// V_WMMA_F32_16X16X4_F32 (matches the reference's f32 matmul numerics).
//
// Block = 8 waves; each wave owns a 16-row Z panel held in registers. The 16x64
// E tile (4 KB) is staged into double-buffered LDS once per block per K-step
// (8x cut in L2 traffic), software-pipelined with one barrier per iteration.
// B panel is bulk-loaded from LDS into registers (clause of ds_load_2addr_b64,
// incremental dscnt waits) before the two interleaved WMMA accumulator chains.

typedef __attribute__((ext_vector_type(2))) float v2f;
typedef __attribute__((ext_vector_type(8))) float v8f;

#define DIM   64
#define LROW  68   // padded LDS row stride (f32 words): conflict-free b64 reads
#define WAVES 8    // waves (16-row tiles) per block

// ---- Pass 1: precompute ||e_k||^2 into workspace -------------------------
__global__ __launch_bounds__(256) void enorm_kernel(const float* __restrict__ emb,
                                                    float* __restrict__ enorm, int K) {
    int k = blockIdx.x * blockDim.x + threadIdx.x;
    if (k >= K) return;
    const float4* p = (const float4*)(emb + (size_t)k * DIM);
    float s = 0.f;
#pragma unroll
    for (int i = 0; i < DIM / 4; ++i) {
        float4 v = p[i];
        s = fmaf(v.x, v.x, s);
        s = fmaf(v.y, v.y, s);
        s = fmaf(v.z, v.z, s);
        s = fmaf(v.w, v.w, s);
    }
    enorm[k] = s;
}

// ---- Pass 2: LDS-staged WMMA distance tiles + running argmin -------------
// A (16x4 f32) per lane l: row = l&15, k-pair base = 2*(l>>4).
// C (16x16 f32): vgpr r of lane l -> (M = r + 8*(l>>4), Ncol = l&15).
__global__ __launch_bounds__(256) void vq_argmin_kernel(const float* __restrict__ z,
                                                        const float* __restrict__ emb,
                                                        const float* __restrict__ enorm,
                                                        int* __restrict__ out,
                                                        int N, int K) {
    __shared__ float lds[2][16 * LROW];

    const int tid  = threadIdx.x;
    const int lane = tid & 31;
    const int wave = tid >> 5;

    int row0 = (blockIdx.x * WAVES + wave) * 16;
    const bool active = row0 < N;          // wave-uniform; inactive waves still
    if (!active) row0 = 0;                 // run (barriers + EXEC-all-1 WMMA)

    const int half = lane >> 4;
    const int l16  = lane & 15;

    // Preload the 16x64 A panel (reused across all K tiles): 16 v2f regs.
    v2f a[DIM / 4];
    {
        const float* zrow = z + (size_t)(row0 + l16) * DIM + 2 * half;
#pragma unroll
        for (int s = 0; s < DIM / 4; ++s) a[s] = *(const v2f*)(zrow + 4 * s);
    }

    // Cooperative staging coordinates: thread -> (row, 4-col) of the E tile.
    const int sAddr = (tid >> 4) * LROW + (tid & 15) * 4;      // LDS store word
    const float* gsrc = emb + (size_t)(tid >> 4) * DIM + (tid & 15) * 4;

    // Pipeline prologue: fetch tile 0 into registers.
    float4 stage = *(const float4*)gsrc;

    float best[8];
    int   bidx[8];
#pragma unroll
    for (int r = 0; r < 8; ++r) { best[r] = __builtin_inff(); bidx[r] = 0; }

    const int ldsBase = l16 * LROW + 2 * half;

    for (int col0 = 0; col0 < K; col0 += 16) {
        const int buf = (col0 >> 4) & 1;

        // Publish tile col0 to LDS; one barrier/iter is safe with double buffer.
        *(float4*)&lds[buf][sAddr] = stage;
        __syncthreads();

        // Issue long-latency ops early: next tile fetch + this tile's ||e||^2.
        const int nxt = (col0 + 16 < K) ? (col0 + 16) : 0;
        stage = *(const float4*)(gsrc + (size_t)nxt * DIM);
        const float en = enorm[col0 + l16];

        // Bulk-load the whole B panel (16 x v2f) -> one DS clause, then WMMAs
        // consume with incremental dscnt waits.
        v2f b[DIM / 4];
#pragma unroll
        for (int s = 0; s < DIM / 4; ++s)
            b[s] = *(const v2f*)&lds[buf][ldsBase + 4 * s];

        // 16 WMMA k-steps, two independent accumulator chains.
        v8f c0 = {0.f, 0.f, 0.f, 0.f, 0.f, 0.f, 0.f, 0.f};
        v8f c1 = {0.f, 0.f, 0.f, 0.f, 0.f, 0.f, 0.f, 0.f};
#pragma unroll
        for (int s = 0; s < DIM / 4; s += 2) {
            c0 = __builtin_amdgcn_wmma_f32_16x16x4_f32(false, a[s],     false, b[s],
                                                       (short)0, c0, false, false);
            c1 = __builtin_amdgcn_wmma_f32_16x16x4_f32(false, a[s + 1], false, b[s + 1],
                                                       (short)0, c1, false, false);
        }

        const int n = col0 + l16;
#pragma unroll
        for (int r = 0; r < 8; ++r) {
            float score = fmaf(-2.0f, c0[r] + c1[r], en);  // ||e||^2 - 2 z.e
            bool take   = score < best[r];                 // strict < => first-min
            best[r] = take ? score : best[r];
            bidx[r] = take ? n : bidx[r];
        }
    }

    // Argmin reduction across the 16 lanes (columns) of each half, with
    // smallest-index tie-breaking to match jnp.argmin.
#pragma unroll
    for (int off = 1; off < 16; off <<= 1) {
#pragma unroll
        for (int r = 0; r < 8; ++r) {
            float ob = __shfl_xor(best[r], off, 32);
            int   oi = __shfl_xor(bidx[r], off, 32);
            bool take = (ob < best[r]) || (ob == best[r] && oi < bidx[r]);
            best[r] = take ? ob : best[r];
            bidx[r] = take ? oi : bidx[r];
        }
    }

    if (active && l16 == 0) {
#pragma unroll
        for (int r = 0; r < 8; ++r)
            out[row0 + r + 8 * half] = bidx[r];
    }
}

extern "C" void kernel_launch(void* const* d_in, const int* in_sizes, int n_in,
                              void* d_out, int out_size, void* d_ws, size_t ws_size,
                              hipStream_t stream) {
    const float* z   = (const float*)d_in[0];
    const float* gw  = (const float*)d_in[1];  // g > 0: argmin-invariant, unused
    const float* emb = (const float*)d_in[2];
    (void)gw; (void)n_in; (void)ws_size; (void)out_size;

    const int N = in_sizes[1];            // g_weight has N elements
    const int K = in_sizes[2] / DIM;      // embedding is K x 64

    float* enorm = (float*)d_ws;          // K floats of scratch

    enorm_kernel<<<(K + 255) / 256, 256, 0, stream>>>(emb, enorm, K);

    const int tiles  = (N + 15) / 16;                    // 16-row tiles, one per wave
    const int blocks = (tiles + WAVES - 1) / WAVES;      // 8 waves (256 thr) per block
    vq_argmin_kernel<<<blocks, 256, 0, stream>>>(z, emb, enorm, (int*)d_out, N, K);
}